// SparseMoE_63067299774601
// MI455X (gfx1250) — compile-verified
//
#include <hip/hip_runtime.h>
#include <hip/hip_bf16.h>
#include <math.h>

// Problem constants (match reference)
#define NB 4
#define NS 2048
#define NTOK (NB * NS)      // 8192 tokens
#define DD 768
#define FF 3072
#define NE 8

// Tiling
#define M_TOK 32            // tokens per block
#define TILES 256           // NTOK / M_TOK (worst case all tokens -> one expert)
#define FC 64               // F chunk
#define KB 32               // K block for W1 staging
#define NKB (DD / KB)       // 24 K-blocks per chunk
#define NCHUNK (FF / FC)    // 48 chunks
#define NSTAGE (NCHUNK * NKB)
#define XS_STRIDE 772       // 768 + 4  -> bank-friendly A reads
#define W1S_STRIDE 72       // 64 + 8   -> bank-friendly B reads
#define HS_STRIDE 68        // 64 + 4   -> bank-friendly A reads

typedef float v2f __attribute__((ext_vector_type(2)));
typedef float v8f __attribute__((ext_vector_type(8)));
typedef int v4i_t __attribute__((vector_size(16)));   // matches builtin param

#if __has_builtin(__builtin_amdgcn_global_load_async_to_lds_b128)
#define HAS_ASYNC_LDS 1
#else
#define HAS_ASYNC_LDS 0
#endif

__device__ __forceinline__ void copy16_to_lds(const float* g, float* l) {
#if HAS_ASYNC_LDS
  __builtin_amdgcn_global_load_async_to_lds_b128(
      (__attribute__((address_space(1))) v4i_t*)g,
      (__attribute__((address_space(3))) v4i_t*)l, 0, 0);
#else
  *(float4*)l = *(const float4*)g;
#endif
}

__device__ __forceinline__ void stage_wait_barrier() {
#if HAS_ASYNC_LDS
#if __has_builtin(__builtin_amdgcn_s_wait_asynccnt)
  __builtin_amdgcn_s_wait_asynccnt(0);
#else
  asm volatile("s_wait_asynccnt 0" ::: "memory");
#endif
#endif
  __syncthreads();
}

__device__ __forceinline__ v8f wmma_f32(v2f a, v2f b, v8f c) {
  // D = A(16x4, f32) * B(4x16, f32) + C(16x16, f32)
  return __builtin_amdgcn_wmma_f32_16x16x4_f32(false, a, false, b, (short)0, c,
                                               false, false);
}

// ---------------------------------------------------------------------------
// Router: noisy top-2 gating; builds per-expert (token, gate) lists.
// ---------------------------------------------------------------------------
__global__ __launch_bounds__(256) void moe_router_kernel(
    const float* __restrict__ x, const float* __restrict__ noise,
    const float* __restrict__ Wg, const float* __restrict__ bg,
    const float* __restrict__ Wn, const float* __restrict__ bn,
    int* __restrict__ counts, int* __restrict__ tokIdx,
    float* __restrict__ tokGate) {
  const int t = blockIdx.x * blockDim.x + threadIdx.x;
  if (t >= NTOK) return;

  float lg[NE], nl[NE];
#pragma unroll
  for (int e = 0; e < NE; ++e) { lg[e] = bg[e]; nl[e] = bn[e]; }

  const float4* xv = (const float4*)(x + (size_t)t * DD);
  for (int d4 = 0; d4 < DD / 4; ++d4) {
    float4 v = xv[d4];
    float xa[4] = {v.x, v.y, v.z, v.w};
    const float* wg = Wg + (size_t)d4 * 4 * NE;
    const float* wn = Wn + (size_t)d4 * 4 * NE;
#pragma unroll
    for (int dd = 0; dd < 4; ++dd) {
#pragma unroll
      for (int e = 0; e < NE; ++e) {
        lg[e] = fmaf(xa[dd], wg[dd * NE + e], lg[e]);
        nl[e] = fmaf(xa[dd], wn[dd * NE + e], nl[e]);
      }
    }
  }

  float noisy[NE];
#pragma unroll
  for (int e = 0; e < NE; ++e) {
    float s = nl[e];
    float sp = (s > 30.f) ? s : log1pf(expf(s));   // softplus
    noisy[e] = lg[e] + noise[(size_t)t * NE + e] * sp;
  }

  // top-2, jax tie-break (lowest index wins on ties -> strict >, ascending)
  int i0 = 0; float v0 = noisy[0];
#pragma unroll
  for (int e = 1; e < NE; ++e) if (noisy[e] > v0) { v0 = noisy[e]; i0 = e; }
  int i1 = (i0 == 0) ? 1 : 0; float v1 = noisy[i1];
#pragma unroll
  for (int e = 0; e < NE; ++e)
    if (e != i0 && noisy[e] > v1) { v1 = noisy[e]; i1 = e; }

  float ex = expf(v1 - v0);           // softmax over the two selected logits
  float inv = 1.f / (1.f + ex);
  float g0 = inv, g1 = ex * inv;

  int s0 = atomicAdd(&counts[i0], 1);
  tokIdx[i0 * NTOK + s0] = t; tokGate[i0 * NTOK + s0] = g0;
  int s1 = atomicAdd(&counts[i1], 1);
  tokIdx[i1 * NTOK + s1] = t; tokGate[i1 * NTOK + s1] = g1;
}

// ---------------------------------------------------------------------------
// Fused expert MLP: out[tok] += gate * ( relu(x @ W1[e] + b1[e]) @ W2[e] + b2[e] )
// Block = 32 gathered tokens of one expert, 8 waves. FP32 WMMA (16x16x4).
// W1 staged into double-buffered LDS via async LDS loads (when available).
// ---------------------------------------------------------------------------
__global__ __launch_bounds__(256) void moe_expert_kernel(
    const float* __restrict__ x, const int* __restrict__ tokIdx,
    const float* __restrict__ tokGate, const int* __restrict__ counts,
    const float* __restrict__ W1, const float* __restrict__ b1,
    const float* __restrict__ W2, const float* __restrict__ b2,
    float* __restrict__ out) {
  extern __shared__ float smem[];
  float* xs  = smem;                           // [32][XS_STRIDE]  x tile
  float* w1s = xs + M_TOK * XS_STRIDE;         // 2x [32][W1S_STRIDE] W1 K-block
  float* hs  = w1s + 2 * KB * W1S_STRIDE;      // [32][HS_STRIDE]  relu(H) chunk

  const int e    = blockIdx.x / TILES;
  const int tile = blockIdx.x % TILES;
  const int cnt  = counts[e];
  const int base = tile * M_TOK;
  if (base >= cnt) return;

  const int tid  = threadIdx.x;
  const int lane = tid & 31;
  const int wave = tid >> 5;
  const int ln16 = lane & 15;
  const int half = lane >> 4;

  const float* W1e = W1 + (size_t)e * DD * FF;
  const float* W2e = W2 + (size_t)e * FF * DD;

  // ---- gather 32 token rows into LDS (coalesced 128B per 8 lanes) ----
  {
    const int r = tid >> 3, seg = tid & 7;
    const int slot = base + r;
    const int tok = (slot < cnt) ? tokIdx[e * NTOK + slot]
                                 : tokIdx[e * NTOK + base];   // defined pad
    const float* src = x + (size_t)tok * DD;
    float* dst = xs + r * XS_STRIDE;
#pragma unroll 4
    for (int i = 0; i < 24; ++i) {
      int c = (seg + i * 8) * 4;               // float index, 16B aligned
      copy16_to_lds(src + c, dst + c);
    }
  }

  // ---- stage 0 of W1 pipeline ----
  {
    const int row = tid >> 4, q = (tid & 15) * 4;          // 16 rows x 16 quads
#pragma unroll
    for (int i = 0; i < 2; ++i)
      copy16_to_lds(W1e + (size_t)(row + 16 * i) * FF + q,
                    w1s + (row + 16 * i) * W1S_STRIDE + q);
  }

  v8f acc2[6][2];
#pragma unroll
  for (int t = 0; t < 6; ++t) { acc2[t][0] = (v8f){}; acc2[t][1] = (v8f){}; }

  const int nT  = wave & 3;                    // GEMM1 f-tile within chunk
  const int mT1 = wave >> 2;                   // GEMM1 m-tile
  const int dColBase = wave * 96;              // GEMM2: wave owns D cols [w*96, w*96+96)
  const int srow = tid >> 4;                   // staging row (0..15)
  const int sq   = (tid & 15) * 4;             // staging float4 col

  int gs = 0;                                  // linear W1 stage counter
  for (int fc = 0; fc < FF; fc += FC) {
    // prefetch this chunk's W2 rows (64 x 768 f32 = 1536 cachelines)
    {
      const float* w2c = W2e + (size_t)fc * DD;
#pragma unroll
      for (int i = 0; i < 6; ++i)
        __builtin_prefetch(w2c + (size_t)(tid + 256 * i) * 32, 0, 1);
    }
    // -------- GEMM1: Hc[32,64] = X[32,768] @ W1e[:, fc:fc+64] --------
    v8f acc1 = (v8f){};
    for (int kbi = 0; kbi < NKB; ++kbi) {
      stage_wait_barrier();                    // stage gs ready in buf[gs&1]
      if (gs + 1 < NSTAGE) {                   // issue stage gs+1 into other buf
        const int s2  = gs + 1;
        const int fc2 = (s2 / NKB) * FC;
        const int kb2 = (s2 % NKB) * KB;
        float* buf2 = w1s + (s2 & 1) * (KB * W1S_STRIDE);
#pragma unroll
        for (int i = 0; i < 2; ++i)
          copy16_to_lds(W1e + (size_t)(kb2 + srow + 16 * i) * FF + fc2 + sq,
                        buf2 + (srow + 16 * i) * W1S_STRIDE + sq);
      }
      const float* wbuf = w1s + (gs & 1) * (KB * W1S_STRIDE);
      const float* xrow =
          xs + (mT1 * 16 + ln16) * XS_STRIDE + kbi * KB + half * 2;
#pragma unroll
      for (int ks = 0; ks < KB / 4; ++ks) {    // K=4 per WMMA
        v2f a, b;
        a.x = xrow[ks * 4];  a.y = xrow[ks * 4 + 1];
        const float* wrow = wbuf + (ks * 4 + half * 2) * W1S_STRIDE + nT * 16 + ln16;
        b.x = wrow[0];       b.y = wrow[W1S_STRIDE];
        acc1 = wmma_f32(a, b, acc1);
      }
      ++gs;
    }
    // -------- bias + ReLU -> hs --------
    __syncthreads();                           // prev chunk's hs readers done
    {
      const int fcol = nT * 16 + ln16;
      const float bb = b1[(size_t)e * FF + fc + fcol];
#pragma unroll
      for (int j = 0; j < 8; ++j) {
        int m = mT1 * 16 + j + half * 8;       // C layout: M = j + 8*half
        float h = acc1[j] + bb;
        hs[m * HS_STRIDE + fcol] = h > 0.f ? h : 0.f;
      }
    }
    __syncthreads();
    // -------- GEMM2: Y[32, w*96 .. +96) += Hc @ W2e[fc:fc+64, :] --------
#pragma unroll 4
    for (int ks = 0; ks < FC / 4; ++ks) {
      const int k0 = ks * 4 + half * 2;
      v2f a0, a1;
      a0.x = hs[ln16 * HS_STRIDE + k0];        a0.y = hs[ln16 * HS_STRIDE + k0 + 1];
      a1.x = hs[(16 + ln16) * HS_STRIDE + k0]; a1.y = hs[(16 + ln16) * HS_STRIDE + k0 + 1];
      const float* w2p = W2e + (size_t)(fc + k0) * DD + dColBase + ln16;
#pragma unroll
      for (int t = 0; t < 6; ++t) {
        v2f b;
        b.x = w2p[t * 16];
        b.y = w2p[t * 16 + DD];
        acc2[t][0] = wmma_f32(a0, b, acc2[t][0]);
        acc2[t][1] = wmma_f32(a1, b, acc2[t][1]);
      }
    }
  }

  // -------- epilogue: out[tok] += gate * (Y + b2) (2 commutative adds/elem) --------
#pragma unroll
  for (int mT = 0; mT < 2; ++mT) {
#pragma unroll
    for (int j = 0; j < 8; ++j) {
      const int m = mT * 16 + j + half * 8;
      const int slot = base + m;
      if (slot < cnt) {
        const int tok = tokIdx[e * NTOK + slot];
        const float g = tokGate[e * NTOK + slot];
#pragma unroll
        for (int t = 0; t < 6; ++t) {
          const int col = dColBase + t * 16 + ln16;
          float v = g * (acc2[t][mT][j] + b2[(size_t)e * DD + col]);
          atomicAdd(out + (size_t)tok * DD + col, v);
        }
      }
    }
  }
}

// ---------------------------------------------------------------------------
extern "C" void kernel_launch(void* const* d_in, const int* in_sizes, int n_in,
                              void* d_out, int out_size, void* d_ws,
                              size_t ws_size, hipStream_t stream) {
  const float* x     = (const float*)d_in[0];
  const float* noise = (const float*)d_in[1];
  const float* Wg    = (const float*)d_in[2];
  const float* bg    = (const float*)d_in[3];
  const float* Wn    = (const float*)d_in[4];
  const float* bn    = (const float*)d_in[5];
  const float* W1    = (const float*)d_in[6];
  const float* b1    = (const float*)d_in[7];
  const float* W2    = (const float*)d_in[8];
  const float* b2    = (const float*)d_in[9];
  float* out = (float*)d_out;

  // workspace layout
  int*   counts  = (int*)d_ws;                                  // E ints
  int*   tokIdx  = (int*)((char*)d_ws + 256);                   // E*NTOK ints
  float* tokGate = (float*)((char*)d_ws + 256 + (size_t)NE * NTOK * 4);

  (void)hipMemsetAsync(d_out, 0, (size_t)out_size * sizeof(float), stream);
  (void)hipMemsetAsync(counts, 0, NE * sizeof(int), stream);

  moe_router_kernel<<<NTOK / 256, 256, 0, stream>>>(x, noise, Wg, bg, Wn, bn,
                                                    counts, tokIdx, tokGate);

  const size_t shmem =
      (size_t)(M_TOK * XS_STRIDE + 2 * KB * W1S_STRIDE + M_TOK * HS_STRIDE) *
      sizeof(float);
  moe_expert_kernel<<<NE * TILES, 256, shmem, stream>>>(
      x, tokIdx, tokGate, counts, W1, b1, W2, b2, out);
}